// graph_emb_9663676416454
// MI455X (gfx1250) — compile-verified
//
#include <hip/hip_runtime.h>
#include <hip/hip_bf16.h>

typedef float v2f __attribute__((ext_vector_type(2)));
typedef float v8f __attribute__((ext_vector_type(8)));

#define D 128

// ---------------- degree / dinv ----------------
__global__ void init_deg_kernel(float* deg, int n) {
    int i = blockIdx.x * blockDim.x + threadIdx.x;
    if (i < n) deg[i] = 1.0f;   // self loop
}

__global__ void accum_deg_kernel(const int* __restrict__ dst, float* deg, int e) {
    int i = blockIdx.x * blockDim.x + threadIdx.x;
    if (i < e) atomicAdd(&deg[dst[i]], 1.0f);
}

__global__ void rsqrt_kernel(float* deg, int n) {
    int i = blockIdx.x * blockDim.x + threadIdx.x;
    if (i < n) deg[i] = rsqrtf(deg[i]);
}

// ---------------- fp32 WMMA GEMM: Y[m][n] = sum_k X[m][k] * W[n][k] ----------------
// One wave computes a 16(M) x 128(N) stripe with V_WMMA_F32_16X16X4_F32.
// Per k-step: 1 A frag + 8 B frags loaded into DISTINCT regs (single wait point),
// then 8 back-to-back WMMAs on independent accumulator chains.
// 64 acc VGPRs + ~2x18 frag VGPRs: no spills, good occupancy.
// A 16x4 layout : lane L holds A[L&15][2*(L>>4)+j] in VGPR j
// B 4x16 layout : lane L holds B[2*(L>>4)+j][L&15] in VGPR j
// C/D 16x16     : lane L, VGPR r holds C[r + 8*(L>>4)][L&15]
__global__ void __launch_bounds__(256)
gemm_xwt_wmma(const float* __restrict__ X, const float* __restrict__ W,
              float* __restrict__ Y, int nrows) {
    const int lane = threadIdx.x & 31;
    const int wave = blockIdx.x * (blockDim.x >> 5) + (threadIdx.x >> 5);
    const int row0 = wave * 16;
    if (row0 + 16 > nrows) return;          // uniform per-wave guard, EXEC stays full

    const int half = lane >> 4;             // 0 or 1
    const int lm   = lane & 15;

    v8f zero = {};
    v8f acc[8];
#pragma unroll
    for (int t = 0; t < 8; ++t) acc[t] = zero;

    const float* xrow = X + (size_t)(row0 + lm) * D;
    const float* wrow = W + (size_t)lm * D;

#pragma unroll 2
    for (int k0 = 0; k0 < D; k0 += 4) {
        const int ka = k0 + 2 * half;
        v2f a = *(const v2f*)(xrow + ka);
        v2f b[8];
#pragma unroll
        for (int t = 0; t < 8; ++t)
            b[t] = *(const v2f*)(wrow + (size_t)(t * 16) * D + ka);   // B[k][n] = W[n][k]
#pragma unroll
        for (int t = 0; t < 8; ++t)
            acc[t] = __builtin_amdgcn_wmma_f32_16x16x4_f32(
                         false, a, false, b[t], (short)0, acc[t], false, false);
    }

#pragma unroll
    for (int t = 0; t < 8; ++t) {
#pragma unroll
        for (int r = 0; r < 8; ++r) {
            Y[(size_t)(row0 + r + 8 * half) * D + t * 16 + lm] = acc[t][r];
        }
    }
}

// scalar tail for rows not covered by 16-row tiles (unused when N % 16 == 0)
__global__ void gemm_xwt_tail(const float* __restrict__ X, const float* __restrict__ W,
                              float* __restrict__ Y, int row_start, int nrows) {
    int idx = blockIdx.x * blockDim.x + threadIdx.x;
    int row = row_start + (idx >> 7);
    int n = idx & (D - 1);
    if (row >= nrows) return;
    const float* xr = X + (size_t)row * D;
    const float* wr = W + (size_t)n * D;
    float s = 0.f;
#pragma unroll 8
    for (int k = 0; k < D; ++k) s = fmaf(xr[k], wr[k], s);
    Y[(size_t)row * D + n] = s;
}

// ---------------- agg init: agg = xw*dinv^2 + b (+ optional base), float4 ----------------
__global__ void agg_init_kernel(const float4* __restrict__ xw, const float* __restrict__ dinv,
                                const float4* __restrict__ b, const float4* __restrict__ base,
                                float4* __restrict__ out, int n4) {
    int idx = blockIdx.x * blockDim.x + threadIdx.x;
    if (idx >= n4) return;
    int node = idx >> 5;                    // 32 float4 per 128-wide row
    int f4 = idx & 31;
    float dv = dinv[node];
    float d2 = dv * dv;
    float4 x = xw[idx];
    float4 bb = b[f4];
    float4 v;
    v.x = fmaf(x.x, d2, bb.x);
    v.y = fmaf(x.y, d2, bb.y);
    v.z = fmaf(x.z, d2, bb.z);
    v.w = fmaf(x.w, d2, bb.w);
    if (base) {
        float4 r = base[idx];
        v.x += r.x; v.y += r.y; v.z += r.z; v.w += r.w;
    }
    out[idx] = v;
}

// ---------------- edge scatter: agg[dst] += xw[src]*dinv[src]*dinv[dst] ----------------
// one wave per edge (wave-uniform edge index -> scalar loads);
// lane handles 4 consecutive features (float4 coalesced gather + 4 f32 atomics)
__global__ void edge_scatter_kernel(const float* __restrict__ xw, const float* __restrict__ dinv,
                                    const int* __restrict__ src, const int* __restrict__ dst,
                                    float* __restrict__ agg, int nedges) {
    int e = blockIdx.x * (blockDim.x >> 5) + (threadIdx.x >> 5);  // wave-uniform
    if (e >= nedges) return;
    int lane = threadIdx.x & 31;
    int s = src[e];
    int d = dst[e];
    float norm = dinv[s] * dinv[d];
    const float4 v = *(const float4*)(xw + (size_t)s * D + lane * 4);
    float* p = agg + (size_t)d * D + lane * 4;
    atomicAdd(p + 0, v.x * norm);
    atomicAdd(p + 1, v.y * norm);
    atomicAdd(p + 2, v.z * norm);
    atomicAdd(p + 3, v.w * norm);
}

// ---------------- epilogue: out = relu(agg) + res, float4 ----------------
__global__ void relu_residual_kernel(const float4* __restrict__ agg, const float4* __restrict__ res,
                                     float4* __restrict__ out, int n4) {
    int idx = blockIdx.x * blockDim.x + threadIdx.x;
    if (idx >= n4) return;
    float4 a = agg[idx];
    float4 r = res[idx];
    float4 o;
    o.x = fmaxf(a.x, 0.0f) + r.x;
    o.y = fmaxf(a.y, 0.0f) + r.y;
    o.z = fmaxf(a.z, 0.0f) + r.z;
    o.w = fmaxf(a.w, 0.0f) + r.w;
    out[idx] = o;
}

extern "C" void kernel_launch(void* const* d_in, const int* in_sizes, int n_in,
                              void* d_out, int out_size, void* d_ws, size_t ws_size,
                              hipStream_t stream) {
    const float* X  = (const float*)d_in[0];
    const int*   EI = (const int*)d_in[1];
    const float* W1 = (const float*)d_in[2];
    const float* b1 = (const float*)d_in[3];
    const float* W2 = (const float*)d_in[4];
    const float* b2 = (const float*)d_in[5];
    float* out = (float*)d_out;

    const int N  = in_sizes[0] / D;
    const int E  = in_sizes[1] / 2;
    const int NF = N * D;
    const int NF4 = NF / 4;
    const int* src = EI;
    const int* dst = EI + E;

    float* ws   = (float*)d_ws;
    float* dinv = ws;               // N floats
    float* xw   = ws + 65536;       // NF floats
    float* agg  = xw + NF;          // NF floats
    float* hbuf = agg + NF;         // NF floats

    const int T = 256;
    const int gN   = (N + T - 1) / T;
    const int gE   = (E + T - 1) / T;
    const int gNF4 = (NF4 + T - 1) / T;
    const int gSC  = (E + 7) / 8;               // 8 waves (edges) per 256-thr block

    const int full_tiles = N / 16;
    const int gGEMM = (full_tiles + 7) / 8;     // 8 waves (16-row tiles) per block
    const int rem_start = full_tiles * 16;
    const int rem = N - rem_start;

    // degree -> dinv (same for all layers)
    init_deg_kernel<<<gN, T, 0, stream>>>(dinv, N);
    accum_deg_kernel<<<gE, T, 0, stream>>>(dst, dinv, E);
    rsqrt_kernel<<<gN, T, 0, stream>>>(dinv, N);

    const float4* b1v = (const float4*)b1;
    const float4* b2v = (const float4*)b2;

    // ---- layer 1: h2 = relu(conv(x, W1, b1)) + x ----
    gemm_xwt_wmma<<<gGEMM, T, 0, stream>>>(X, W1, xw, N);
    if (rem > 0)
        gemm_xwt_tail<<<(rem * D + T - 1) / T, T, 0, stream>>>(X, W1, xw, rem_start, N);
    agg_init_kernel<<<gNF4, T, 0, stream>>>((const float4*)xw, dinv, b1v, nullptr,
                                            (float4*)agg, NF4);
    edge_scatter_kernel<<<gSC, T, 0, stream>>>(xw, dinv, src, dst, agg, E);
    relu_residual_kernel<<<gNF4, T, 0, stream>>>((const float4*)agg, (const float4*)X,
                                                 (float4*)hbuf, NF4);   // hbuf = h2

    // ---- layer 2: h3 = relu(conv(h2, W2, b2)) + h2 ----
    gemm_xwt_wmma<<<gGEMM, T, 0, stream>>>(hbuf, W2, xw, N);
    if (rem > 0)
        gemm_xwt_tail<<<(rem * D + T - 1) / T, T, 0, stream>>>(hbuf, W2, xw, rem_start, N);
    agg_init_kernel<<<gNF4, T, 0, stream>>>((const float4*)xw, dinv, b2v, nullptr,
                                            (float4*)agg, NF4);
    edge_scatter_kernel<<<gSC, T, 0, stream>>>(xw, dinv, src, dst, agg, E);
    relu_residual_kernel<<<gNF4, T, 0, stream>>>((const float4*)agg, (const float4*)hbuf,
                                                 (float4*)hbuf, NF4);   // hbuf = h3 (in place)

    // ---- layer 3: out = conv(h3, W2, b2) + h3 (no relu) ----
    gemm_xwt_wmma<<<gGEMM, T, 0, stream>>>(hbuf, W2, xw, N);
    if (rem > 0)
        gemm_xwt_tail<<<(rem * D + T - 1) / T, T, 0, stream>>>(hbuf, W2, xw, rem_start, N);
    agg_init_kernel<<<gNF4, T, 0, stream>>>((const float4*)xw, dinv, b2v, (const float4*)hbuf,
                                            (float4*)out, NF4);          // out = xw*dinv^2+b+h3
    edge_scatter_kernel<<<gSC, T, 0, stream>>>(xw, dinv, src, dst, out, E);
}